// NbrAgg_29051158790654
// MI455X (gfx1250) — compile-verified
//
#include <hip/hip_runtime.h>

typedef __attribute__((ext_vector_type(2))) float v2f;
typedef __attribute__((ext_vector_type(8))) float v8f;

#define PTS_B 2
#define PTS_N 8192
#define NPTS  (PTS_B * PTS_N)     // 16384
#define KNN   16
#define NROWS (NPTS * KNN)        // 262144
#define EPS_BN 1e-5f

// ---- workspace layout (float offsets) ----
#define OFF_FEAT 0u                         // NROWS*8
#define OFF_Y1   2097152u                   // NROWS*16
#define OFF_Y2   6291456u                   // NROWS*64
#define OFF_Y3   23068672u                  // NPTS*64
#define OFF_Z    24117248u                  // NPTS*128
#define OFF_Y4   26214400u                  // NPTS*64
#define OFF_ST   27262976u                  // 512 (st1@0 len32, st2@32 len128, st3@160 len128, st4@288 len128)
#define OFF_CF   27263488u                  // 512 (cf1@0 len32, cf2@32 len128, cf3@160 len128, cf4@288 len128)

// =====================================================================
// K1: brute-force KNN (top-17 incl self) + edge feature build
// one thread = one query point; candidates staged through LDS
// =====================================================================
__global__ __launch_bounds__(256) void knn_feat_kernel(
    const float* __restrict__ pts, float* __restrict__ feat) {
  __shared__ float spx[256], spy[256], spz[256];
  const int tid = threadIdx.x;
  const int gid = blockIdx.x * 256 + tid;      // global point index
  const int b   = gid / PTS_N;
  const float* bp = pts + (long)b * PTS_N * 3;

  const float qx = pts[(long)gid * 3 + 0];
  const float qy = pts[(long)gid * 3 + 1];
  const float qz = pts[(long)gid * 3 + 2];

  float bd[17]; int bi[17];
  int cnt = 0, wslot = 0;
  float worst = -1.0e30f;

  for (int chunk = 0; chunk < PTS_N; chunk += 256) {
    const int j = chunk + tid;
    spx[tid] = bp[(long)j * 3 + 0];
    spy[tid] = bp[(long)j * 3 + 1];
    spz[tid] = bp[(long)j * 3 + 2];
    __syncthreads();
    for (int t = 0; t < 256; ++t) {
      const float dx = qx - spx[t], dy = qy - spy[t], dz = qz - spz[t];
      const float d  = dx * dx + dy * dy + dz * dz;
      const int   cj = chunk + t;
      if (cnt < 17) {
        bd[cnt] = d; bi[cnt] = cj;
        if (d > worst) { worst = d; wslot = cnt; }
        cnt++;
      } else if (d < worst) {
        bd[wslot] = d; bi[wslot] = cj;
        worst = bd[0]; wslot = 0;
        for (int u = 1; u < 17; ++u)
          if (bd[u] > worst) { worst = bd[u]; wslot = u; }
      }
    }
    __syncthreads();
  }

  // sort ascending by (dist, index) -- matches top_k(-d2) ordering
  for (int a = 0; a < 16; ++a) {
    int m = a;
    for (int u = a + 1; u < 17; ++u)
      if (bd[u] < bd[m] || (bd[u] == bd[m] && bi[u] < bi[m])) m = u;
    float td = bd[a]; bd[a] = bd[m]; bd[m] = td;
    int   ti = bi[a]; bi[a] = bi[m]; bi[m] = ti;
  }

  const float ax = bp[(long)bi[0] * 3 + 0];
  const float ay = bp[(long)bi[0] * 3 + 1];
  const float az = bp[(long)bi[0] * 3 + 2];
  for (int k = 0; k < KNN; ++k) {
    const int j = bi[k + 1];
    const float rx = bp[(long)j * 3 + 0] - ax;
    const float ry = bp[(long)j * 3 + 1] - ay;
    const float rz = bp[(long)j * 3 + 2] - az;
    const float dist = sqrtf(rx * rx + ry * ry + rz * rz + 1e-8f);
    float* f = feat + ((long)gid * KNN + k) * 8;
    f[0] = ax; f[1] = ay; f[2] = az;
    f[3] = rx; f[4] = ry; f[5] = rz;
    f[6] = dist; f[7] = 0.0f;
  }
}

// =====================================================================
// K2: pts lift  y3 = pts @ W2^T  (16384 x 64)  + BN3 stats
// =====================================================================
__global__ __launch_bounds__(256) void lift_kernel(
    const float* __restrict__ pts, const float* __restrict__ W2,
    float* __restrict__ y3, float* __restrict__ st3) {
  __shared__ float ls[64], lq[64];
  const int tid = threadIdx.x;
  if (tid < 64) { ls[tid] = 0.0f; lq[tid] = 0.0f; }
  __syncthreads();
  const int gid = blockIdx.x * 256 + tid;
  const int p = gid >> 6, c = gid & 63;
  const float v = pts[(long)p * 3 + 0] * W2[c * 3 + 0] +
                  pts[(long)p * 3 + 1] * W2[c * 3 + 1] +
                  pts[(long)p * 3 + 2] * W2[c * 3 + 2];
  y3[gid] = v;
  atomicAdd(&ls[c], v);
  atomicAdd(&lq[c], v * v);
  __syncthreads();
  if (tid < 64) { atomicAdd(&st3[tid], ls[tid]); atomicAdd(&st3[64 + tid], lq[tid]); }
}

// =====================================================================
// K3: GEMM1 via v_wmma_f32_16x16x4_f32
//     y1(262144x16) = feat(262144x8pad) @ W1a^T(8x16)   + BN1 stats
// one wave = one 16-row tile; 2 K-steps of 4
// =====================================================================
__global__ __launch_bounds__(256) void gemm1_kernel(
    const float* __restrict__ feat, const float* __restrict__ W1a,
    float* __restrict__ y1, float* __restrict__ st1) {
  __shared__ float ls[16], lq[16];
  const int tid = threadIdx.x;
  if (tid < 16) { ls[tid] = 0.0f; lq[tid] = 0.0f; }
  __syncthreads();
  const int lane = tid & 31, wave = tid >> 5;
  const int row0 = (blockIdx.x * 8 + wave) * 16;
  const int m     = lane & 15;           // A row / B,D column within tile
  const int koff  = (lane >> 4) * 2;     // K sub-offset per half-wave
  const int mbase = (lane >> 4) * 8;     // D row base per half-wave
  const int row   = row0 + m;

  v8f c = {};
#pragma unroll
  for (int s = 0; s < 2; ++s) {
    const int k0 = s * 4 + koff;
    v2f a, bm;
    a.x = feat[(long)row * 8 + k0];
    a.y = feat[(long)row * 8 + k0 + 1];
    bm.x = (k0     < 7) ? W1a[m * 7 + k0    ] : 0.0f;   // B[k][n] = W1a[n][k]
    bm.y = (k0 + 1 < 7) ? W1a[m * 7 + k0 + 1] : 0.0f;
    c = __builtin_amdgcn_wmma_f32_16x16x4_f32(false, a, false, bm,
                                              (short)0, c, false, false);
  }
  float s = 0.0f, q = 0.0f;
#pragma unroll
  for (int v = 0; v < 8; ++v) {
    const float val = c[v];
    y1[(long)(row0 + mbase + v) * 16 + m] = val;
    s += val; q += val * val;
  }
  atomicAdd(&ls[m], s);
  atomicAdd(&lq[m], q);
  __syncthreads();
  if (tid < 16) { atomicAdd(&st1[tid], ls[tid]); atomicAdd(&st1[16 + tid], lq[tid]); }
}

// =====================================================================
// K4: BN finalize -> per-channel scale/shift
// =====================================================================
__global__ void bn_finalize_kernel(const float* __restrict__ st,
                                   const float* __restrict__ gamma,
                                   const float* __restrict__ beta,
                                   float* __restrict__ coef, int C, float invCnt) {
  const int c = threadIdx.x;
  if (c < C) {
    const float mean = st[c] * invCnt;
    const float var  = st[C + c] * invCnt - mean * mean;
    const float sc   = gamma[c] * rsqrtf(var + EPS_BN);
    coef[c]     = sc;
    coef[C + c] = beta[c] - mean * sc;
  }
}

// =====================================================================
// K5: GEMM2 with fused BN1+ReLU on A
//     y2(262144x64) = relu(bn1(y1)) @ W1b^T(16x64)   + BN2 stats
// one wave = 16 rows x 64 cols (4 N-tiles), 4 K-steps
// =====================================================================
__global__ __launch_bounds__(256) void gemm2_kernel(
    const float* __restrict__ y1, const float* __restrict__ coef1,
    const float* __restrict__ W1b, float* __restrict__ y2,
    float* __restrict__ st2) {
  __shared__ float ls[64], lq[64];
  const int tid = threadIdx.x;
  if (tid < 64) { ls[tid] = 0.0f; lq[tid] = 0.0f; }
  __syncthreads();
  const int lane = tid & 31, wave = tid >> 5;
  const int row0 = (blockIdx.x * 8 + wave) * 16;
  const int m     = lane & 15;
  const int koff  = (lane >> 4) * 2;
  const int mbase = (lane >> 4) * 8;
  const int row   = row0 + m;

  v8f acc[4] = {};
#pragma unroll
  for (int s = 0; s < 4; ++s) {
    const int k0 = s * 4 + koff;
    v2f a;
    const float a0 = y1[(long)row * 16 + k0    ] * coef1[k0    ] + coef1[16 + k0    ];
    const float a1 = y1[(long)row * 16 + k0 + 1] * coef1[k0 + 1] + coef1[16 + k0 + 1];
    a.x = fmaxf(a0, 0.0f);
    a.y = fmaxf(a1, 0.0f);
#pragma unroll
    for (int t = 0; t < 4; ++t) {
      const int n = t * 16 + m;
      v2f bm;
      bm.x = W1b[n * 16 + k0];
      bm.y = W1b[n * 16 + k0 + 1];
      acc[t] = __builtin_amdgcn_wmma_f32_16x16x4_f32(false, a, false, bm,
                                                     (short)0, acc[t], false, false);
    }
  }
#pragma unroll
  for (int t = 0; t < 4; ++t) {
    const int n = t * 16 + m;
    float s = 0.0f, q = 0.0f;
#pragma unroll
    for (int v = 0; v < 8; ++v) {
      const float val = acc[t][v];
      y2[(long)(row0 + mbase + v) * 64 + n] = val;
      s += val; q += val * val;
    }
    atomicAdd(&ls[n], s);
    atomicAdd(&lq[n], q);
  }
  __syncthreads();
  if (tid < 64) { atomicAdd(&st2[tid], ls[tid]); atomicAdd(&st2[64 + tid], lq[tid]); }
}

// =====================================================================
// K6: apply BN2+ReLU then max-pool over K; concat with BN3+ReLU lift
//     z(16384x128) = [relu(bn3(y3)) | maxpool_k relu(bn2(y2))]
// =====================================================================
__global__ __launch_bounds__(256) void pool_concat_kernel(
    const float* __restrict__ y2, const float* __restrict__ coef2,
    const float* __restrict__ y3, const float* __restrict__ coef3,
    float* __restrict__ z) {
  const int gid = blockIdx.x * 256 + threadIdx.x;
  const int p = gid >> 7, cc = gid & 127;
  float out;
  if (cc < 64) {
    out = fmaxf(y3[(long)p * 64 + cc] * coef3[cc] + coef3[64 + cc], 0.0f);
  } else {
    const int c = cc - 64;
    const float sc = coef2[c], sh = coef2[64 + c];
    float mx = 0.0f;   // relu outputs are >= 0
    for (int k = 0; k < KNN; ++k) {
      const float v = fmaxf(y2[((long)p * KNN + k) * 64 + c] * sc + sh, 0.0f);
      mx = fmaxf(mx, v);
    }
    out = mx;
  }
  z[gid] = out;
}

// =====================================================================
// K7: GEMM4  y4(16384x64) = z(16384x128) @ W3^T(128x64)  + BN4 stats
// one wave = 16 rows x 64 cols, 32 K-steps
// =====================================================================
__global__ __launch_bounds__(256) void gemm4_kernel(
    const float* __restrict__ z, const float* __restrict__ W3,
    float* __restrict__ y4, float* __restrict__ st4) {
  __shared__ float ls[64], lq[64];
  const int tid = threadIdx.x;
  if (tid < 64) { ls[tid] = 0.0f; lq[tid] = 0.0f; }
  __syncthreads();
  const int lane = tid & 31, wave = tid >> 5;
  const int row0 = (blockIdx.x * 8 + wave) * 16;
  const int m     = lane & 15;
  const int koff  = (lane >> 4) * 2;
  const int mbase = (lane >> 4) * 8;
  const int row   = row0 + m;

  v8f acc[4] = {};
#pragma unroll
  for (int s = 0; s < 32; ++s) {
    const int k0 = s * 4 + koff;
    v2f a;
    a.x = z[(long)row * 128 + k0];
    a.y = z[(long)row * 128 + k0 + 1];
#pragma unroll
    for (int t = 0; t < 4; ++t) {
      const int n = t * 16 + m;
      v2f bm;
      bm.x = W3[n * 128 + k0];
      bm.y = W3[n * 128 + k0 + 1];
      acc[t] = __builtin_amdgcn_wmma_f32_16x16x4_f32(false, a, false, bm,
                                                     (short)0, acc[t], false, false);
    }
  }
#pragma unroll
  for (int t = 0; t < 4; ++t) {
    const int n = t * 16 + m;
    float s = 0.0f, q = 0.0f;
#pragma unroll
    for (int v = 0; v < 8; ++v) {
      const float val = acc[t][v];
      y4[(long)(row0 + mbase + v) * 64 + n] = val;
      s += val; q += val * val;
    }
    atomicAdd(&ls[n], s);
    atomicAdd(&lq[n], q);
  }
  __syncthreads();
  if (tid < 64) { atomicAdd(&st4[tid], ls[tid]); atomicAdd(&st4[64 + tid], lq[tid]); }
}

// =====================================================================
// K8: final BN4 + ReLU -> d_out
// =====================================================================
__global__ __launch_bounds__(256) void out_kernel(
    const float* __restrict__ y4, const float* __restrict__ coef4,
    float* __restrict__ out) {
  const int gid = blockIdx.x * 256 + threadIdx.x;
  const int c = gid & 63;
  out[gid] = fmaxf(y4[gid] * coef4[c] + coef4[64 + c], 0.0f);
}

// =====================================================================
extern "C" void kernel_launch(void* const* d_in, const int* in_sizes, int n_in,
                              void* d_out, int out_size, void* d_ws, size_t ws_size,
                              hipStream_t stream) {
  (void)in_sizes; (void)n_in; (void)out_size; (void)ws_size;
  const float* pts = (const float*)d_in[0];
  const float* W1a = (const float*)d_in[1];
  const float* g1a = (const float*)d_in[2];
  const float* b1a = (const float*)d_in[3];
  const float* W1b = (const float*)d_in[4];
  const float* g1b = (const float*)d_in[5];
  const float* b1b = (const float*)d_in[6];
  const float* W2  = (const float*)d_in[7];
  const float* g2  = (const float*)d_in[8];
  const float* b2  = (const float*)d_in[9];
  const float* W3  = (const float*)d_in[10];
  const float* g3  = (const float*)d_in[11];
  const float* b3  = (const float*)d_in[12];

  float* ws   = (float*)d_ws;
  float* feat = ws + OFF_FEAT;
  float* y1   = ws + OFF_Y1;
  float* y2   = ws + OFF_Y2;
  float* y3   = ws + OFF_Y3;
  float* z    = ws + OFF_Z;
  float* y4   = ws + OFF_Y4;
  float* st1  = ws + OFF_ST;          // 32
  float* st2  = ws + OFF_ST + 32;     // 128
  float* st3  = ws + OFF_ST + 160;    // 128
  float* st4  = ws + OFF_ST + 288;    // 128
  float* cf1  = ws + OFF_CF;          // 32
  float* cf2  = ws + OFF_CF + 32;     // 128
  float* cf3  = ws + OFF_CF + 160;    // 128
  float* cf4  = ws + OFF_CF + 288;    // 128

  // zero the stat accumulators every launch (ws is poisoned, never re-zeroed)
  hipMemsetAsync(ws + OFF_ST, 0, 512 * sizeof(float), stream);

  knn_feat_kernel<<<NPTS / 256, 256, 0, stream>>>(pts, feat);
  lift_kernel<<<(NPTS * 64) / 256, 256, 0, stream>>>(pts, W2, y3, st3);
  gemm1_kernel<<<NROWS / 128, 256, 0, stream>>>(feat, W1a, y1, st1);
  bn_finalize_kernel<<<1, 64, 0, stream>>>(st1, g1a, b1a, cf1, 16, 1.0f / (float)NROWS);
  gemm2_kernel<<<NROWS / 128, 256, 0, stream>>>(y1, cf1, W1b, y2, st2);
  bn_finalize_kernel<<<1, 64, 0, stream>>>(st2, g1b, b1b, cf2, 64, 1.0f / (float)NROWS);
  bn_finalize_kernel<<<1, 64, 0, stream>>>(st3, g2, b2, cf3, 64, 1.0f / (float)NPTS);
  pool_concat_kernel<<<(NPTS * 128) / 256, 256, 0, stream>>>(y2, cf2, y3, cf3, z);
  gemm4_kernel<<<NPTS / 128, 256, 0, stream>>>(z, W3, y4, st4);
  bn_finalize_kernel<<<1, 64, 0, stream>>>(st4, g3, b3, cf4, 64, 1.0f / (float)NPTS);
  out_kernel<<<(NPTS * 64) / 256, 256, 0, stream>>>(y4, cf4, (float*)d_out);
}